// FullyConnectedPredictor_30245159698934
// MI455X (gfx1250) — compile-verified
//
#include <hip/hip_runtime.h>

// ---------------------------------------------------------------------------
// FullyConnectedPredictor on MI455X (gfx1250)
//
// Exact linear rewrite:
//   score[e,c] = 10*(g[src,c]+g[dst,c]) + edge_feat[e,:] . M[:,c] + c0[c]
//   g  = h @ W2[0:128,:]          (50000 x 2,  WMMA f32 16x16x4, N padded to 16)
//   M  = W1 @ W2[128:256,:]       (32 x 2)
//   c0 = b1 @ W2[128:256,:] + b2  (2)
// HBM traffic: ~1.16 GB naive -> ~170 MB (edge_feat streamed NT once; g-table
// 400 KB lives in the 192 MB L2). Roofline ~7 us at 23.3 TB/s.
// ---------------------------------------------------------------------------

typedef float v2f __attribute__((ext_vector_type(2)));
typedef float v4f __attribute__((ext_vector_type(4)));
typedef float v8f __attribute__((ext_vector_type(8)));

#define N_NODES 50000
#define N_EDGES 1000000

// workspace layout (float offsets)
#define WS_M0   0        // M column 0  [32]
#define WS_M1   32       // M column 1  [32]
#define WS_C0   64       // c0          [2]
#define WS_BP   128      // Bpad, per-lane WMMA order: [32 steps][32 lanes][2]
#define WS_G    2176     // g           [N_NODES*2]
// total: 2176 + 100000 floats ~= 409 KB

// --- tiny precompute: M = W1 @ W2b, c0 = b1 @ W2b + b2, Bpad ---------------
__global__ __launch_bounds__(64)
void fcp_prep_kernel(const float* __restrict__ W1, const float* __restrict__ b1,
                     const float* __restrict__ W2, const float* __restrict__ b2,
                     float* __restrict__ ws) {
  int t = threadIdx.x;          // 0..63
  int c = t >> 5;               // output column 0/1
  int j = t & 31;               // edge-feature index
  float acc = 0.f;
  for (int k = 0; k < 128; ++k)
    acc += W1[j * 128 + k] * W2[(128 + k) * 2 + c];
  ws[(c ? WS_M1 : WS_M0) + j] = acc;
  if (j == 0) {
    float cc = b2[c];
    for (int k = 0; k < 128; ++k)
      cc += b1[k] * W2[(128 + k) * 2 + c];
    ws[WS_C0 + c] = cc;
  }
  // Bpad[i]: i = step*64 + lane*2 + comp.  Lane 'lane' of the gmat wave loads
  // the v2f at (step*32 + lane): { B[k0][n], B[k0+1][n] } with
  // n = lane&15, k0 = 4*step + 2*(lane>>4);  B[k][n] = (n<2) ? W2[k*2+n] : 0.
  for (int i = t; i < 32 * 32 * 2; i += 64) {
    int step = i >> 6;
    int rem  = i & 63;
    int lane = rem >> 1;
    int comp = rem & 1;
    int n    = lane & 15;
    int k    = step * 4 + ((lane >> 4) << 1) + comp;
    ws[WS_BP + i] = (n < 2) ? W2[k * 2 + n] : 0.0f;
  }
}

// --- g = h @ W2a via V_WMMA_F32_16X16X4_F32 ---------------------------------
// One wave per 16-node tile. A: 16x4 f32 (lane%16 = M row; lane/16 selects the
// K-half; the 2 A VGPRs are consecutive K). B comes pre-swizzled from Bpad so
// the inner loop is branch-free: 2x global_load_b64 + v_wmma per K-step.
__global__ __launch_bounds__(256)
void fcp_gmat_kernel(const float* __restrict__ h, float* __restrict__ ws) {
  const int NT = N_NODES / 16;                       // 3125 tiles exactly
  int wid  = (blockIdx.x * blockDim.x + threadIdx.x) >> 5;
  if (wid >= NT) return;                             // wave-uniform branch
  int lane  = threadIdx.x & 31;
  int row   = lane & 15;                             // A: M row / store column
  int khalf = (lane >> 4) << 1;                      // 0 or 2
  const float* hrow = h + (size_t)(wid * 16 + row) * 128 + khalf;
  const v2f* bp = reinterpret_cast<const v2f*>(ws + WS_BP) + lane;  // +32/step
  v8f c = {0.f, 0.f, 0.f, 0.f, 0.f, 0.f, 0.f, 0.f};
#pragma unroll 8
  for (int kk = 0; kk < 32; ++kk) {
    v2f a = *reinterpret_cast<const v2f*>(hrow + kk * 4);
    v2f b = bp[kk * 32];
    c = __builtin_amdgcn_wmma_f32_16x16x4_f32(false, a, false, b,
                                              (short)0, c, false, false);
  }
  if (row < 2) {                                     // only N<2 columns real
    int mbase = (lane >> 4) << 3;
    float* g = ws + WS_G;
#pragma unroll
    for (int r = 0; r < 8; ++r)
      g[(size_t)(wid * 16 + mbase + r) * 2 + row] = c[r];
  }
}

// --- streaming edge kernel: 1 thread = 1 edge -------------------------------
__global__ __launch_bounds__(256)
void fcp_edge_kernel(const float* __restrict__ ef,
                     const int* __restrict__ src, const int* __restrict__ dst,
                     const float* __restrict__ ws, float* __restrict__ out) {
  int e = blockIdx.x * blockDim.x + threadIdx.x;
  if (e >= N_EDGES) return;
  const v4f* a  = reinterpret_cast<const v4f*>(ef + (size_t)e * 32);
  const v4f* M0 = reinterpret_cast<const v4f*>(ws + WS_M0);
  const v4f* M1 = reinterpret_cast<const v4f*>(ws + WS_M1);
  float acc0 = ws[WS_C0 + 0];
  float acc1 = ws[WS_C0 + 1];
#pragma unroll
  for (int jj = 0; jj < 8; ++jj) {
    v4f av = __builtin_nontemporal_load(&a[jj]);     // 128 MB read-once stream
    v4f p  = M0[jj];                                 // L0-broadcast hits
    v4f q  = M1[jj];
    acc0 = fmaf(av.x, p.x, acc0); acc0 = fmaf(av.y, p.y, acc0);
    acc0 = fmaf(av.z, p.z, acc0); acc0 = fmaf(av.w, p.w, acc0);
    acc1 = fmaf(av.x, q.x, acc1); acc1 = fmaf(av.y, q.y, acc1);
    acc1 = fmaf(av.z, q.z, acc1); acc1 = fmaf(av.w, q.w, acc1);
  }
  int s = src[e], d = dst[e];
  const v2f* g = reinterpret_cast<const v2f*>(ws + WS_G);
  v2f gs = g[s];                                     // 400 KB table: L2-resident
  v2f gd = g[d];
  v2f r;
  r.x = acc0 + 10.f * (gs.x + gd.x);
  r.y = acc1 + 10.f * (gs.y + gd.y);
  __builtin_nontemporal_store(r, reinterpret_cast<v2f*>(out + (size_t)e * 2));
}

extern "C" void kernel_launch(void* const* d_in, const int* in_sizes, int n_in,
                              void* d_out, int out_size, void* d_ws, size_t ws_size,
                              hipStream_t stream) {
  const float* h   = (const float*)d_in[0];
  const float* ef  = (const float*)d_in[1];
  const float* W1  = (const float*)d_in[2];
  const float* b1  = (const float*)d_in[3];
  const float* W2  = (const float*)d_in[4];
  const float* b2  = (const float*)d_in[5];
  const int*   src = (const int*)d_in[6];
  const int*   dst = (const int*)d_in[7];
  float* out = (float*)d_out;
  float* ws  = (float*)d_ws;   // ~409 KB used; fully rewritten every call

  fcp_prep_kernel<<<1, 64, 0, stream>>>(W1, b1, W2, b2, ws);

  int ntiles  = N_NODES / 16;                 // 3125
  int gblocks = (ntiles + 7) / 8;             // 8 waves (tiles) per 256-thr block
  fcp_gmat_kernel<<<gblocks, 256, 0, stream>>>(h, ws);

  int eblocks = (N_EDGES + 255) / 256;
  fcp_edge_kernel<<<eblocks, 256, 0, stream>>>(ef, src, dst, ws, out);
}